// GNN_8452495639089
// MI455X (gfx1250) — compile-verified
//
#include <hip/hip_runtime.h>
#include <hip/hip_bf16.h>
#include <cstdint>

// ---------------------------------------------------------------------------
// GAT (2-layer, single head) for MI455X / gfx1250, wave32.
//   Layer: h = x@W ; s_src = h@a_src ; s_dst = h@a_dst
//          e = leaky_relu(s_src[src]+s_dst[dst]) ; softmax over dst segments
//          out[v] = sum alpha * h[src] + b  (+ relu after layer 1)
// GEMMs use V_WMMA_F32_16X16X4_F32 (exact fp32). W is staged in LDS in a
// K-pair-interleaved layout so every B fragment is one aligned ds_load_b64
// directly into the even-aligned VGPR pair WMMA wants (no mov shuffles).
// Aggregation is CSR-based, wave-per-destination-node, lanes across feature
// dims -> coalesced 256B gathers of h[src], no float atomics anywhere.
// ---------------------------------------------------------------------------

typedef __attribute__((ext_vector_type(2))) float v2f;
typedef __attribute__((ext_vector_type(8))) float v8f;

#define N_NODES 50000
#define N_EDGES 800000
#define IN_DIM  128
#define HID_DIM 64

// ------------------------------- utility -----------------------------------
__global__ void k_fill_i32(int* __restrict__ p, int n, int val) {
    int i = blockIdx.x * blockDim.x + threadIdx.x;
    if (i < n) p[i] = val;
}

// ---------------------------- CSR construction ------------------------------
// Virtual edge stream: [0, nE) real edges, [nE, nE+nNodes) self loops.
__global__ void k_count(const int* __restrict__ ei, int nE, int nNodes,
                        int* __restrict__ counts) {
    int i = blockIdx.x * blockDim.x + threadIdx.x;
    int tot = nE + nNodes;
    if (i >= tot) return;
    int dst = (i < nE) ? ei[nE + i] : (i - nE);
    atomicAdd(&counts[dst], 1);
}

// Single-block (1024 threads) exclusive scan -> offsets[0..n], offsets[0]=0.
__global__ void k_scan(const int* __restrict__ counts, int n,
                       int* __restrict__ offsets) {
    __shared__ int buf[1024];
    __shared__ int carry;
    int tid = threadIdx.x;
    if (tid == 0) { carry = 0; offsets[0] = 0; }
    __syncthreads();
    for (int base = 0; base < n; base += 1024) {
        int i = base + tid;
        int v = (i < n) ? counts[i] : 0;
        buf[tid] = v;
        __syncthreads();
        // Hillis-Steele inclusive scan
        for (int off = 1; off < 1024; off <<= 1) {
            int t = (tid >= off) ? buf[tid - off] : 0;
            __syncthreads();
            buf[tid] += t;
            __syncthreads();
        }
        int c = carry;
        if (i < n) offsets[i + 1] = c + buf[tid];
        __syncthreads();
        if (tid == 0) carry = c + buf[1023];
        __syncthreads();
    }
}

__global__ void k_scatter(const int* __restrict__ ei, int nE, int nNodes,
                          const int* __restrict__ offsets,
                          int* __restrict__ cursor, int* __restrict__ srcs) {
    int i = blockIdx.x * blockDim.x + threadIdx.x;
    int tot = nE + nNodes;
    if (i >= tot) return;
    int src, dst;
    if (i < nE) { src = ei[i]; dst = ei[nE + i]; }
    else        { src = dst = i - nE; }
    int pos = offsets[dst] + atomicAdd(&cursor[dst], 1);
    srcs[pos] = src;
}

// ----------------------- fused GEMM + attention scores ----------------------
// Block = 256 threads = 8 waves. Wave w handles node rows [m0, m0+16).
// Each wave: 16x64 output tile via 4x v8f WMMA accumulators, K-loop step 4.
// W staged in LDS K-pair-interleaved: sW[(k/2)*128 + 2*col] = {W[k][col],
// W[k+1][col]} so each B fragment is one aligned 8B LDS load. The LDS buffer
// is then reused to linearize the accumulator tile for coalesced h stores +
// fused s_src/s_dst dot products.
template <int KDIM>
__global__ __launch_bounds__(256) void k_gemm_scores(
    const float* __restrict__ X,     // [nNodes, KDIM]
    const float* __restrict__ W,     // [KDIM, 64] row-major
    const float* __restrict__ a_src, // [64]
    const float* __restrict__ a_dst, // [64]
    float* __restrict__ H,           // [nNodes, 64]
    float* __restrict__ Ssrc,        // [nNodes]
    float* __restrict__ Sdst,        // [nNodes]
    int nNodes) {
    constexpr int WELEMS = KDIM * 64;      // packed W uses same float count
    constexpr int HELEMS = 8 * 16 * 64;    // 8 waves x (16x64) tile
    constexpr int CAP = (WELEMS > HELEMS) ? WELEMS : HELEMS;
    __shared__ __align__(16) float sBuf[CAP]; // phase 1: packed W ; phase 2: h
    __shared__ float sA[128];                 // a_src | a_dst

    const int tid  = threadIdx.x;
    const int wave = tid >> 5;
    const int lane = tid & 31;
    const int half = lane >> 4;   // 0: lanes 0-15, 1: lanes 16-31
    const int l16  = lane & 15;
    const int m0   = (blockIdx.x * 8 + wave) * 16;

    // Stage W K-pair-interleaved: pair p, col c -> {W[2p][c], W[2p+1][c]}
    for (int i = tid; i < (KDIM / 2) * 64; i += 256) {
        int p = i >> 6, c = i & 63;
        sBuf[i * 2 + 0] = W[(2 * p + 0) * 64 + c];
        sBuf[i * 2 + 1] = W[(2 * p + 1) * 64 + c];
    }
    for (int i = tid; i < 64; i += 256) { sA[i] = a_src[i]; sA[64 + i] = a_dst[i]; }
    __syncthreads();

    // A fragment row for this lane (clamped; stores are guarded later)
    int arow = m0 + l16;
    if (arow >= nNodes) arow = nNodes - 1;
    const float* __restrict__ xrow = X + (size_t)arow * KDIM + half * 2;

    v8f acc0 = {}, acc1 = {}, acc2 = {}, acc3 = {};
    for (int k = 0; k < KDIM; k += 4) {
        // A 16x4 fragment: v0 = K(0|2), v1 = K(1|3) across lane halves
        v2f a = *(const v2f*)(xrow + k);
        // B fragments: one aligned 8B LDS word each
        const float* __restrict__ bbase =
            &sBuf[((k >> 1) + half) * 128 + 2 * l16];
#pragma unroll
        for (int n = 0; n < 4; ++n) {
            v2f b = *(const v2f*)(bbase + n * 32);
            v8f c = (n == 0) ? acc0 : (n == 1) ? acc1 : (n == 2) ? acc2 : acc3;
            c = __builtin_amdgcn_wmma_f32_16x16x4_f32(
                    false, a, false, b, (short)0, c, false, false);
            if (n == 0) acc0 = c; else if (n == 1) acc1 = c;
            else if (n == 2) acc2 = c; else acc3 = c;
        }
    }

    __syncthreads();  // all waves done reading W from sBuf
    // Linearize C layout (VGPR r: lanes0-15 -> row r, lanes16-31 -> row r+8)
    float* __restrict__ hB = sBuf + wave * (16 * 64);
#pragma unroll
    for (int r = 0; r < 8; ++r) {
        int row = r + half * 8;
        hB[row * 64 +  0 + l16] = acc0[r];
        hB[row * 64 + 16 + l16] = acc1[r];
        hB[row * 64 + 32 + l16] = acc2[r];
        hB[row * 64 + 48 + l16] = acc3[r];
    }
    __syncthreads();

    // Coalesced h stores: each lane writes cols [2*lane, 2*lane+1]
#pragma unroll
    for (int r = 0; r < 16; ++r) {
        int row = m0 + r;
        if (row < nNodes) {
            float2 hv;
            hv.x = hB[r * 64 + 2 * lane];
            hv.y = hB[r * 64 + 2 * lane + 1];
            *(float2*)(H + (size_t)row * 64 + 2 * lane) = hv;
        }
    }

    // Fused attention scores: lanes 0-15 -> s_src of rows 0..15,
    // lanes 16-31 -> s_dst of rows 0..15.
    {
        const float* __restrict__ av = sA + half * 64;
        const float* __restrict__ hr = hB + l16 * 64;
        float s = 0.f;
#pragma unroll
        for (int d = 0; d < 64; ++d) s += hr[d] * av[d];
        int row = m0 + l16;
        if (row < nNodes) {
            if (half) Sdst[row] = s; else Ssrc[row] = s;
        }
    }
}

// ------------------- softmax + aggregation (wave per node) ------------------
// Lanes stripe the 64 feature dims (2/lane) -> each h[src] gather is one
// coalesced 256B wavefront load. Max pass is lane-parallel over edges,
// accumulate pass is serial over edges (avg degree ~17).
__global__ __launch_bounds__(256) void k_aggregate(
    const int* __restrict__ offsets, const int* __restrict__ srcs,
    const float* __restrict__ H, const float* __restrict__ Ssrc,
    const float* __restrict__ Sdst, const float* __restrict__ bias,
    float* __restrict__ Out, int nNodes, int doRelu) {
    int gwave = (blockIdx.x * blockDim.x + threadIdx.x) >> 5;
    int lane = threadIdx.x & 31;
    if (gwave >= nNodes) return;
    const int v = gwave;
    const int beg = offsets[v], end = offsets[v + 1];
    const float sdv = Sdst[v];

    // pass 1: segment max (lane-parallel over edges)
    float m = -3.0e38f;
    for (int j = beg + lane; j < end; j += 32) {
        float e = Ssrc[srcs[j]] + sdv;
        e = (e >= 0.f) ? e : 0.2f * e;
        m = fmaxf(m, e);
    }
#pragma unroll
    for (int off = 16; off > 0; off >>= 1) m = fmaxf(m, __shfl_xor(m, off, 32));

    // pass 2: exp/sum + weighted feature gather
    float acc0 = 0.f, acc1 = 0.f, denom = 0.f;
    const int d0 = 2 * lane;
    for (int j = beg; j < end; ++j) {
        int s = srcs[j];
        if (j + 1 < end)
            __builtin_prefetch(H + (size_t)srcs[j + 1] * 64, 0, 1);
        float e = Ssrc[s] + sdv;                  // broadcast scalar-ish load
        e = (e >= 0.f) ? e : 0.2f * e;
        float p = __expf(e - m);
        denom += p;
        float2 hv = *(const float2*)(H + (size_t)s * 64 + d0);
        acc0 += p * hv.x;
        acc1 += p * hv.y;
    }
    float inv = 1.0f / denom;
    float o0 = acc0 * inv + bias[d0];
    float o1 = acc1 * inv + bias[d0 + 1];
    if (doRelu) { o0 = fmaxf(o0, 0.f); o1 = fmaxf(o1, 0.f); }
    float2 ov; ov.x = o0; ov.y = o1;
    *(float2*)(Out + (size_t)v * 64 + d0) = ov;
}

// ------------------------------- launcher -----------------------------------
extern "C" void kernel_launch(void* const* d_in, const int* in_sizes, int n_in,
                              void* d_out, int out_size, void* d_ws, size_t ws_size,
                              hipStream_t stream) {
    const float* x     = (const float*)d_in[0];
    const int*   ei    = (const int*)d_in[1];   // [2, E]: row0=src, row1=dst
    const float* W1    = (const float*)d_in[2];
    const float* a1s   = (const float*)d_in[3];
    const float* a1d   = (const float*)d_in[4];
    const float* b1    = (const float*)d_in[5];
    const float* W2    = (const float*)d_in[6];
    const float* a2s   = (const float*)d_in[7];
    const float* a2d   = (const float*)d_in[8];
    const float* b2    = (const float*)d_in[9];
    float* out = (float*)d_out;

    const int N = N_NODES;
    const int E = N_EDGES;
    const int Etot = E + N;  // + self loops

    // Workspace layout (16B aligned slices)
    char* ws = (char*)d_ws;
    size_t off = 0;
    auto take = [&](size_t bytes) {
        void* p = ws + off;
        off += (bytes + 15) & ~size_t(15);
        return p;
    };
    float* bufA    = (float*)take((size_t)N * 64 * sizeof(float)); // h1 / h2
    float* bufB    = (float*)take((size_t)N * 64 * sizeof(float)); // relu(out1)
    float* s_src   = (float*)take((size_t)N * sizeof(float));
    float* s_dst   = (float*)take((size_t)N * sizeof(float));
    int*   counts  = (int*)take((size_t)N * sizeof(int));          // also cursor
    int*   offsets = (int*)take((size_t)(N + 1) * sizeof(int));
    int*   srcs    = (int*)take((size_t)Etot * sizeof(int));
    (void)ws_size;

    const int T = 256;
    dim3 bT(T);
    dim3 gN((N + T - 1) / T);
    dim3 gE((Etot + T - 1) / T);
    dim3 gGemm((N + 127) / 128);          // 8 waves x 16 rows per block
    dim3 gAgg(((size_t)N * 32 + T - 1) / T);

    // --- CSR build (dst-major), includes self loops ---
    k_fill_i32<<<gN, bT, 0, stream>>>(counts, N, 0);
    k_count<<<gE, bT, 0, stream>>>(ei, E, N, counts);
    k_scan<<<1, 1024, 0, stream>>>(counts, N, offsets);
    k_fill_i32<<<gN, bT, 0, stream>>>(counts, N, 0);  // reuse as cursor
    k_scatter<<<gE, bT, 0, stream>>>(ei, E, N, offsets, counts, srcs);

    // --- Layer 1: h1 = x@W1, scores, softmax-aggregate, +b1, relu ---
    k_gemm_scores<IN_DIM><<<gGemm, bT, 0, stream>>>(x, W1, a1s, a1d,
                                                    bufA, s_src, s_dst, N);
    k_aggregate<<<gAgg, bT, 0, stream>>>(offsets, srcs, bufA, s_src, s_dst,
                                         b1, bufB, N, /*relu=*/1);

    // --- Layer 2: h2 = z1@W2, scores, softmax-aggregate, +b2 ---
    k_gemm_scores<HID_DIM><<<gGemm, bT, 0, stream>>>(bufB, W2, a2s, a2d,
                                                     bufA, s_src, s_dst, N);
    k_aggregate<<<gAgg, bT, 0, stream>>>(offsets, srcs, bufA, s_src, s_dst,
                                         b2, out, N, /*relu=*/0);
}